// SRWMlayer_38740605010039
// MI455X (gfx1250) — compile-verified
//
#include <hip/hip_runtime.h>
#include <hip/hip_bf16.h>
#include <math.h>

// ---------------------------------------------------------------------------
// SRWM layer for MI455X (gfx1250, wave32).
//   Kernel 1: sequential self-referential scan, one workgroup (2 waves) per
//             (b,h). Weight state lives in REGISTERS (thread owns its rows);
//             softmax/beta reductions via wave32 shfl butterflies + tiny LDS.
//   Kernel 2: LayerNorm(h) -> d_out (residual base).
//   Kernel 3: fp32 WMMA GEMM  d_out += ys @ out_w^T  using
//             v_wmma_f32_16x16x4_f32 (exact fp32 math, matches reference).
// ---------------------------------------------------------------------------

typedef __attribute__((ext_vector_type(2))) float v2f;
typedef __attribute__((ext_vector_type(8))) float v8f;

#define T_LEN   1024
#define BSZ     32
#define NHEAD   16
#define DHEAD   64
#define IN_DIM  1024           // NHEAD * DHEAD
#define ROWS    (T_LEN * BSZ)  // 32768
#define TSTEP   (BSZ * IN_DIM) // elements per timestep in h / ys

// -------------------------- wave32 reductions ------------------------------
__device__ __forceinline__ float wred_sum(float v) {
    #pragma unroll
    for (int off = 16; off > 0; off >>= 1) v += __shfl_xor(v, off, 32);
    return v;
}
__device__ __forceinline__ float wred_max(float v) {
    #pragma unroll
    for (int off = 16; off > 0; off >>= 1) v = fmaxf(v, __shfl_xor(v, off, 32));
    return v;
}

// ============================ 1. scan kernel ===============================
// grid = 512 blocks (b*16+h), block = 64 threads (2 waves).
// Thread `tid` owns row `tid` of Wy/Wq/Wk (in VGPRs) and row `tid` of wb.
__global__ __launch_bounds__(64)
void srwm_scan_kernel(const float* __restrict__ h,
                      const float* __restrict__ gWy,
                      const float* __restrict__ gWq,
                      const float* __restrict__ gWk,
                      const float* __restrict__ gwb,
                      float* __restrict__ ys)
{
    __shared__ float xs[64];        // broadcast x_t
    __shared__ float eqs[64];       // exp(zq - mq)
    __shared__ float eks[64];       // exp(zk - mk)
    __shared__ float redA[16];      // per-wave partials: my,mq,mk,beta0..3
    __shared__ float redB[24];      // per-wave partials: sy,sq,sk,bk0..3,bq0..3

    const int tid  = threadIdx.x;   // 0..63 : output row o (and input row d)
    const int lane = tid & 31;
    const int wid  = tid >> 5;      // 0 / 1
    const int bh   = blockIdx.x;    // 0..511
    const int b    = bh >> 4;       // 0..31
    const int hh   = bh & 15;       // 0..15

    // ---- weight state in registers: row `tid` of each matrix ----
    float wyr[64], wqr[64], wkr[64], wbr[4];
    {
        const float4* wy4 = (const float4*)(gWy + (size_t)(hh * 64 + tid) * 64);
        const float4* wq4 = (const float4*)(gWq + (size_t)(hh * 64 + tid) * 64);
        const float4* wk4 = (const float4*)(gWk + (size_t)(hh * 64 + tid) * 64);
        #pragma unroll
        for (int q = 0; q < 16; ++q) {
            float4 a = wy4[q]; wyr[4*q] = a.x; wyr[4*q+1] = a.y; wyr[4*q+2] = a.z; wyr[4*q+3] = a.w;
            float4 c = wq4[q]; wqr[4*q] = c.x; wqr[4*q+1] = c.y; wqr[4*q+2] = c.z; wqr[4*q+3] = c.w;
            float4 e = wk4[q]; wkr[4*q] = e.x; wkr[4*q+1] = e.y; wkr[4*q+2] = e.z; wkr[4*q+3] = e.w;
        }
        float4 wb = *(const float4*)(gwb + (size_t)(hh * 64 + tid) * 4);
        wbr[0] = wb.x; wbr[1] = wb.y; wbr[2] = wb.z; wbr[3] = wb.w;
    }

    const size_t xbase = (size_t)b * IN_DIM + hh * DHEAD + tid;
    float xh = h[xbase];                       // x for t=0 (double-buffered)

    for (int t = 0; t < T_LEN; ++t) {
        xs[tid] = xh;
        __syncthreads();                                   // B1
        if (t + 1 < T_LEN) xh = h[xbase + (size_t)(t + 1) * TSTEP];  // prefetch

        // ---- phase 1: y = Wy x, zq = Wq x, zk = Wk x (register W) ----
        float y = 0.0f, zq = 0.0f, zk = 0.0f;
        #pragma unroll
        for (int d = 0; d < 64; ++d) {
            float xv = xs[d];
            y  = fmaf(wyr[d], xv, y);
            zq = fmaf(wqr[d], xv, zq);
            zk = fmaf(wkr[d], xv, zk);
        }
        const float xown = xs[tid];            // this thread's x element

        // ---- group A reductions: maxes + beta dots ----
        float pmy = wred_max(y);
        float pmq = wred_max(zq);
        float pmk = wred_max(zk);
        float pb0 = wred_sum(wbr[0] * xown);
        float pb1 = wred_sum(wbr[1] * xown);
        float pb2 = wred_sum(wbr[2] * xown);
        float pb3 = wred_sum(wbr[3] * xown);
        if (lane == 0) {
            float* r = &redA[wid * 8];
            r[0] = pmy; r[1] = pmq; r[2] = pmk;
            r[3] = pb0; r[4] = pb1; r[5] = pb2; r[6] = pb3;
        }
        __syncthreads();                                   // B2

        const float my = fmaxf(redA[0], redA[8]);
        const float mq = fmaxf(redA[1], redA[9]);
        const float mk = fmaxf(redA[2], redA[10]);
        const float b0 = 1.0f / (1.0f + __expf(-(redA[3] + redA[11])));
        const float b1 = 1.0f / (1.0f + __expf(-(redA[4] + redA[12])));
        const float b2 = 1.0f / (1.0f + __expf(-(redA[5] + redA[13])));
        const float b3 = 1.0f / (1.0f + __expf(-(redA[6] + redA[14])));

        const float eyt = __expf(y  - my);
        const float eqt = __expf(zq - mq);
        const float ekt = __expf(zk - mk);
        eqs[tid] = eqt;
        eks[tid] = ekt;

        // ---- group B reductions: exp sums + bk/bq dots ----
        float psy = wred_sum(eyt);
        float psq = wred_sum(eqt);
        float psk = wred_sum(ekt);
        float pk0 = wred_sum(wbr[0] * ekt);
        float pk1 = wred_sum(wbr[1] * ekt);
        float pk2 = wred_sum(wbr[2] * ekt);
        float pk3 = wred_sum(wbr[3] * ekt);
        float pq0 = wred_sum(wbr[0] * eqt);
        float pq1 = wred_sum(wbr[1] * eqt);
        float pq2 = wred_sum(wbr[2] * eqt);
        float pq3 = wred_sum(wbr[3] * eqt);
        if (lane == 0) {
            float* r = &redB[wid * 12];
            r[0] = psy; r[1] = psq; r[2] = psk;
            r[3] = pk0; r[4] = pk1; r[5] = pk2; r[6] = pk3;
            r[7] = pq0; r[8] = pq1; r[9] = pq2; r[10] = pq3;
        }
        __syncthreads();                                   // B3

        const float isy = 1.0f / (redB[0] + redB[12]);
        const float isq = 1.0f / (redB[1] + redB[13]);
        const float isk = 1.0f / (redB[2] + redB[14]);
        const float bkv0 = (redB[3] + redB[15]) * isk, bqv0 = (redB[7]  + redB[19]) * isq;
        const float bkv1 = (redB[4] + redB[16]) * isk, bqv1 = (redB[8]  + redB[20]) * isq;
        const float bkv2 = (redB[5] + redB[17]) * isk, bqv2 = (redB[9]  + redB[21]) * isq;
        const float bkv3 = (redB[6] + redB[18]) * isk, bqv3 = (redB[10] + redB[22]) * isq;

        // out_t = softmax(y)
        ys[xbase + (size_t)t * TSTEP] = eyt * isy;

        // ---- phase 2: W{y,q,k} @ {k,q} (un-normalized; register W) ----
        float ykn = 0, yqn = 0, qkn = 0, qqn = 0, kkn = 0, kqn = 0;
        #pragma unroll
        for (int d = 0; d < 64; ++d) {
            float ekd = eks[d], eqd = eqs[d];
            ykn = fmaf(wyr[d], ekd, ykn); yqn = fmaf(wyr[d], eqd, yqn);
            qkn = fmaf(wqr[d], ekd, qkn); qqn = fmaf(wqr[d], eqd, qqn);
            kkn = fmaf(wkr[d], ekd, kkn); kqn = fmaf(wkr[d], eqd, kqn);
        }

        // ---- rank-1 updates (all state thread-private) ----
        const float cy = b0 * (yqn * isq - ykn * isk) * isk;
        const float cq = b1 * (qqn * isq - qkn * isk) * isk;
        const float ck = b2 * (kqn * isq - kkn * isk) * isk;
        #pragma unroll
        for (int d = 0; d < 64; ++d) {
            float kd = eks[d];
            wyr[d] = fmaf(cy, kd, wyr[d]);
            wqr[d] = fmaf(cq, kd, wqr[d]);
            wkr[d] = fmaf(ck, kd, wkr[d]);
        }
        const float kdt = ekt * isk;                       // k[tid]
        wbr[0] = fmaf(b3 * kdt, bqv0 - bkv0, wbr[0]);
        wbr[1] = fmaf(b3 * kdt, bqv1 - bkv1, wbr[1]);
        wbr[2] = fmaf(b3 * kdt, bqv2 - bkv2, wbr[2]);
        wbr[3] = fmaf(b3 * kdt, bqv3 - bkv3, wbr[3]);
        // next iteration's B1/B2 order eqs/eks & red* rewrites vs. these reads
    }
}

// ============================ 2. LayerNorm ================================
// grid = 32768 rows, block = 256 threads (4 elems/thread, float4).
__global__ __launch_bounds__(256)
void ln_kernel(const float* __restrict__ h,
               const float* __restrict__ g,
               const float* __restrict__ bb,
               float* __restrict__ out)
{
    __shared__ float red[256];
    const int row = blockIdx.x;
    const int tid = threadIdx.x;
    const float4 v = ((const float4*)(h + (size_t)row * IN_DIM))[tid];

    red[tid] = v.x + v.y + v.z + v.w;
    __syncthreads();
    for (int off = 128; off > 0; off >>= 1) {
        if (tid < off) red[tid] += red[tid + off];
        __syncthreads();
    }
    const float mu = red[0] * (1.0f / IN_DIM);
    __syncthreads();

    const float d0 = v.x - mu, d1 = v.y - mu, d2 = v.z - mu, d3 = v.w - mu;
    red[tid] = d0 * d0 + d1 * d1 + d2 * d2 + d3 * d3;
    __syncthreads();
    for (int off = 128; off > 0; off >>= 1) {
        if (tid < off) red[tid] += red[tid + off];
        __syncthreads();
    }
    const float r = rsqrtf(red[0] * (1.0f / IN_DIM) + 1e-5f);

    const int c = tid * 4;
    const float4 gg = *(const float4*)(g + c);
    const float4 bv = *(const float4*)(bb + c);
    float4 o;
    o.x = d0 * r * gg.x + bv.x;
    o.y = d1 * r * gg.y + bv.y;
    o.z = d2 * r * gg.z + bv.z;
    o.w = d3 * r * gg.w + bv.w;
    ((float4*)(out + (size_t)row * IN_DIM))[tid] = o;
}

// ============================ 3. WMMA GEMM ================================
// out(32768x1024) += ys(32768x1024) @ out_w^T ; out_w is (n,k) row-major so
// B[k][n] = out_w[n*1024+k].  Block tile 128x64, BK=32, 8 waves of 32x32,
// fp32 WMMA 16x16x4 (exact fp32).
#define BM 128
#define BN 64
#define BK 32
#define AST 36   // As row stride: 16B-aligned rows, conflict-free M-reads
#define BST 65   // Bs row stride: conflict-free N-reads

__global__ __launch_bounds__(256)
void gemm_wmma_kernel(const float* __restrict__ A,
                      const float* __restrict__ W,
                      float* __restrict__ out)
{
    __shared__ float As[BM * AST];   // [m][k']  18432 B
    __shared__ float Bs[BK * BST];   // [k'][n]   8320 B

    const int tid   = threadIdx.x;
    const int lane  = tid & 31;
    const int wid   = tid >> 5;      // 0..7
    const int waveM = wid >> 1;      // 0..3 -> 32-row slab
    const int waveN = wid & 1;       // 0..1 -> 32-col slab
    const int l16   = lane & 15;
    const int khalf = lane >> 4;     // 0/1 : K pair select per WMMA layout

    const size_t rowBase = (size_t)blockIdx.x * BM;
    const int    colBase = blockIdx.y * BN;

    // global load assignments
    const int am  = tid >> 1;              // 0..127
    const int ak  = (tid & 1) * 16;        // 0 / 16
    const int bn  = tid >> 2;              // 0..63
    const int bk0 = (tid & 3) * 8;         // 0,8,16,24
    const float* gA = A + (rowBase + am) * IN_DIM + ak;
    const float* gB = W + (size_t)(colBase + bn) * IN_DIM + bk0;

    v8f acc[2][2];
    #pragma unroll
    for (int i = 0; i < 2; ++i)
        #pragma unroll
        for (int j = 0; j < 2; ++j) acc[i][j] = 0.0f;   // splat

    for (int kc = 0; kc < IN_DIM; kc += BK) {
        // ---- stage A chunk: 128x32, 16 floats/thread, float4 LDS stores ----
        {
            float4 a0 = *(const float4*)(gA + kc + 0);
            float4 a1 = *(const float4*)(gA + kc + 4);
            float4 a2 = *(const float4*)(gA + kc + 8);
            float4 a3 = *(const float4*)(gA + kc + 12);
            float* dst = &As[am * AST + ak];
            *(float4*)(dst + 0)  = a0;
            *(float4*)(dst + 4)  = a1;
            *(float4*)(dst + 8)  = a2;
            *(float4*)(dst + 12) = a3;
        }
        // ---- stage B chunk transposed: Bs[k'][n] = out_w[n][kc+k'] ----
        {
            float4 b0 = *(const float4*)(gB + kc + 0);
            float4 b1 = *(const float4*)(gB + kc + 4);
            Bs[(bk0 + 0) * BST + bn] = b0.x;
            Bs[(bk0 + 1) * BST + bn] = b0.y;
            Bs[(bk0 + 2) * BST + bn] = b0.z;
            Bs[(bk0 + 3) * BST + bn] = b0.w;
            Bs[(bk0 + 4) * BST + bn] = b1.x;
            Bs[(bk0 + 5) * BST + bn] = b1.y;
            Bs[(bk0 + 6) * BST + bn] = b1.z;
            Bs[(bk0 + 7) * BST + bn] = b1.w;
        }
        if (kc + BK < IN_DIM) {                 // global_prefetch_b8 next chunk
            __builtin_prefetch(gA + kc + BK, 0, 1);
            __builtin_prefetch(gB + kc + BK, 0, 1);
        }
        __syncthreads();

        // ---- compute: 8 K-steps x 2x2 tiles of v_wmma_f32_16x16x4_f32 ----
        #pragma unroll
        for (int kk = 0; kk < BK; kk += 4) {
            const int kb = kk + 2 * khalf;      // per-lane K base (ISA layout)
            v2f af[2], bf[2];
            #pragma unroll
            for (int i = 0; i < 2; ++i) {
                const int m = waveM * 32 + i * 16 + l16;
                af[i].x = As[m * AST + kb + 0];
                af[i].y = As[m * AST + kb + 1];
            }
            #pragma unroll
            for (int j = 0; j < 2; ++j) {
                const int n = waveN * 32 + j * 16 + l16;
                bf[j].x = Bs[(kb + 0) * BST + n];
                bf[j].y = Bs[(kb + 1) * BST + n];
            }
            #pragma unroll
            for (int i = 0; i < 2; ++i)
                #pragma unroll
                for (int j = 0; j < 2; ++j)
                    acc[i][j] = __builtin_amdgcn_wmma_f32_16x16x4_f32(
                        false, af[i], false, bf[j], 0, acc[i][j], false, false);
        }
        __syncthreads();
    }

    // ---- epilogue: out already holds LN(h); add GEMM result (one writer) ----
    #pragma unroll
    for (int i = 0; i < 2; ++i) {
        #pragma unroll
        for (int j = 0; j < 2; ++j) {
            const int c = colBase + waveN * 32 + j * 16 + l16;
            #pragma unroll
            for (int v = 0; v < 8; ++v) {
                const size_t r = rowBase + waveM * 32 + i * 16 + v + 8 * khalf;
                const size_t idx = r * IN_DIM + c;
                out[idx] = out[idx] + acc[i][j][v];
            }
        }
    }
}

// ============================ launch ======================================
extern "C" void kernel_launch(void* const* d_in, const int* in_sizes, int n_in,
                              void* d_out, int out_size, void* d_ws, size_t ws_size,
                              hipStream_t stream)
{
    const float* h     = (const float*)d_in[0];
    const float* W_y   = (const float*)d_in[1];
    const float* W_q   = (const float*)d_in[2];
    const float* W_k   = (const float*)d_in[3];
    const float* w_b   = (const float*)d_in[4];
    const float* out_w = (const float*)d_in[5];
    const float* ln_g  = (const float*)d_in[6];
    const float* ln_b  = (const float*)d_in[7];
    float* out = (float*)d_out;
    float* ys  = (float*)d_ws;   // needs ROWS*IN_DIM*4 = 128 MiB workspace

    // 1) sequential scan -> ys (workspace)
    srwm_scan_kernel<<<BSZ * NHEAD, 64, 0, stream>>>(h, W_y, W_q, W_k, w_b, ys);
    // 2) LayerNorm(h) -> out (residual base)
    ln_kernel<<<ROWS, 256, 0, stream>>>(h, ln_g, ln_b, out);
    // 3) out += ys @ out_w^T  (fp32 WMMA)
    gemm_wmma_kernel<<<dim3(ROWS / BM, IN_DIM / BN), 256, 0, stream>>>(ys, out_w, out);
}